// RSTDP_57329223467260
// MI455X (gfx1250) — compile-verified
//
#include <hip/hip_runtime.h>

// RSTDP update == rank-1 GER:  out = clip(W + (-0.012*reward) * post_act ⊗ pre_act, -1, 1)
// Memory-bound (512 MB @ 23.3 TB/s ~= 22 us). Mapped onto V_WMMA_F32_16X16X4_F32
// with only K=0 populated (true rank-1 matmul-accumulate), C = weight tile.

typedef __attribute__((ext_vector_type(2))) float v2f;
typedef __attribute__((ext_vector_type(8))) float v8f;

#define N_OUT 8192
#define N_IN  8192
#define A_MINUS 0.012f
#define CTILES 16   // 16x16 column-tiles handled per wave

__global__ __launch_bounds__(256) void rstdp_wmma_rank1(
    const float* __restrict__ W,
    const float* __restrict__ pre_spikes,
    const float* __restrict__ post_spikes,
    const float* __restrict__ reward,
    float* __restrict__ out)
{
    const int lane = threadIdx.x & 31;          // wave32
    const int wave = threadIdx.x >> 5;          // 8 waves / block
    const int row0 = (blockIdx.x * 8 + wave) * 16;      // 16-row strip per wave
    const int col0 = blockIdx.y * (CTILES * 16);

    const float scale = -A_MINUS * reward[0];   // reward * dw factor

    const int  l16     = lane & 15;
    const bool lowHalf = lane < 16;             // K=0 lives in lanes 0..15 of VGPR0

    // A matrix 16x4 f32: VGPR0 = K=0 (lanes 0-15) / K=2 (lanes 16-31); VGPR1 = K=1/K=3.
    // Only K=0 nonzero: a[m,0] = scale if post spike fired for row m.
    float aval = (post_spikes[row0 + l16] > 0.5f) ? scale : 0.0f;
    v2f a; a.x = lowHalf ? aval : 0.0f; a.y = 0.0f;

    // C/D layout: VGPR r <-> row (r + 8*(lane>>4)), col = lane&15
    const int       rOff    = (lane >> 4) * 8;                 // 0 or 8
    const long long rowBase = (long long)(row0 + rOff) * N_IN;

    for (int t = 0; t < CTILES; ++t) {
        const int col = col0 + t * 16;

        // B matrix 4x16 f32, only K=0 row nonzero: b[0,n] = pre_active[n]
        float bval = (pre_spikes[col + l16] > 0.5f) ? 1.0f : 0.0f;
        v2f b; b.x = lowHalf ? bval : 0.0f; b.y = 0.0f;

        // Load weight tile straight into the C/D VGPR layout (8 strided dwords/lane)
        v8f c;
        const float* wp = W + rowBase + col + l16;
        #pragma unroll
        for (int r = 0; r < 8; ++r)
            c[r] = wp[(long long)r * N_IN];

        // D = A x B + C  ->  weight tile + rank-1 outer-product update
        v8f d = __builtin_amdgcn_wmma_f32_16x16x4_f32(
            /*neg_a=*/false, a, /*neg_b=*/false, b,
            /*c_mod=*/(short)0, c, /*reuse_a=*/false, /*reuse_b=*/false);

        // clip to [-1, 1] and stream out through the same layout
        float* op = out + rowBase + col + l16;
        #pragma unroll
        for (int r = 0; r < 8; ++r)
            op[(long long)r * N_IN] = fminf(1.0f, fmaxf(-1.0f, d[r]));
    }
}

extern "C" void kernel_launch(void* const* d_in, const int* in_sizes, int n_in,
                              void* d_out, int out_size, void* d_ws, size_t ws_size,
                              hipStream_t stream) {
    // setup_inputs order: weight, pre_spikes, post_spikes, reward, pre_times, post_times
    const float* W      = (const float*)d_in[0];
    const float* pre_s  = (const float*)d_in[1];
    const float* post_s = (const float*)d_in[2];
    const float* reward = (const float*)d_in[3];
    // d_in[4]/d_in[5] (spike times) cancel out exactly: dts==0 on every masked pair.
    float* out = (float*)d_out;

    // rows: 8192/16 = 512 strips, 8 waves/block -> grid.x = 64
    // cols: 8192 / (CTILES*16) = 32 -> grid.y = 32
    dim3 grid(64, 32);
    rstdp_wmma_rank1<<<grid, 256, 0, stream>>>(W, pre_s, post_s, reward, out);
}